// QGCN_16183436771632
// MI455X (gfx1250) — compile-verified
//
#include <hip/hip_runtime.h>

typedef __attribute__((ext_vector_type(2))) float v2f;
typedef __attribute__((ext_vector_type(8))) float v8f;

#define QMAXF 255.0f

__device__ __forceinline__ v8f wmma_f32_k4(v2f a, v2f b, v8f c) {
  // V_WMMA_F32_16X16X4_F32 : D = A(16x4) * B(4x16) + C(16x16), all f32
  return __builtin_amdgcn_wmma_f32_16x16x4_f32(
      /*neg_a=*/false, a, /*neg_b=*/false, b,
      /*c_mod=*/(short)0, c, /*reuse_a=*/false, /*reuse_b=*/false);
}

// Stochastic quant-dequant of one element.
// Reference: q = clip(round((x-rmin)*rscale + noise - 0.5), 0, 255); dq = q/rscale + rmin
// (inv = 1/rscale precomputed once per row; <=1 ulp from the per-element division)
__device__ __forceinline__ float qdv(float x, float nz, float mn, float scale, float inv) {
  float q = rintf((x - mn) * scale + nz - 0.5f);   // rintf = round-half-even (jnp.round)
  q = fminf(fmaxf(q, 0.0f), QMAXF);
  return q * inv + mn;
}

// ---------------- zero ----------------
__global__ void zero_kernel(float* __restrict__ p, int n) {
  int i = blockIdx.x * blockDim.x + threadIdx.x;
  if (i < n) p[i] = 0.0f;
}

// ---------------- SpMM, d = 16 : z[row[e]] += vals[e] * y[col[e]] ----------------
// 4 lanes per edge, each lane handles 4 contiguous floats (float4 gather).
__global__ void spmm16_kernel(const int* __restrict__ row, const int* __restrict__ col,
                              const float* __restrict__ vals, const float* __restrict__ y,
                              float* __restrict__ z, int nE) {
  int t = blockIdx.x * blockDim.x + threadIdx.x;
  int e = t >> 2;
  if (e >= nE) return;
  int q = (t & 3) << 2;
  int c = col[e];
  int r = row[e];
  float v = vals[e];
  float4 x4 = *reinterpret_cast<const float4*>(y + (size_t)c * 16 + q);
  float* zp = z + (size_t)r * 16 + q;
  __hip_atomic_fetch_add(zp + 0, v * x4.x, __ATOMIC_RELAXED, __HIP_MEMORY_SCOPE_AGENT);
  __hip_atomic_fetch_add(zp + 1, v * x4.y, __ATOMIC_RELAXED, __HIP_MEMORY_SCOPE_AGENT);
  __hip_atomic_fetch_add(zp + 2, v * x4.z, __ATOMIC_RELAXED, __HIP_MEMORY_SCOPE_AGENT);
  __hip_atomic_fetch_add(zp + 3, v * x4.w, __ATOMIC_RELAXED, __HIP_MEMORY_SCOPE_AGENT);
}

// ---------------- standalone quant-dequant, d = 16 (optional ReLU on input) ----------------
template<bool RELU>
__global__ void qd16_kernel(const float* __restrict__ X, const float* __restrict__ NZ,
                            float* __restrict__ O, int n) {
  int r = blockIdx.x * blockDim.x + threadIdx.x;
  if (r >= n) return;
  const float4* xp = reinterpret_cast<const float4*>(X + (size_t)r * 16);
  const float4* np = reinterpret_cast<const float4*>(NZ + (size_t)r * 16);
  float4* op = reinterpret_cast<float4*>(O + (size_t)r * 16);
  float4 xs[4];
  float mn = 3.4e38f, mx = -3.4e38f;
#pragma unroll
  for (int i = 0; i < 4; i++) {
    float4 v = xp[i];
    if (RELU) { v.x = fmaxf(v.x, 0.f); v.y = fmaxf(v.y, 0.f);
                v.z = fmaxf(v.z, 0.f); v.w = fmaxf(v.w, 0.f); }
    xs[i] = v;
    mn = fminf(mn, fminf(fminf(v.x, v.y), fminf(v.z, v.w)));
    mx = fmaxf(mx, fmaxf(fmaxf(v.x, v.y), fmaxf(v.z, v.w)));
  }
  float scale = QMAXF / (mx - mn);
  float inv = 1.0f / scale;
#pragma unroll
  for (int i = 0; i < 4; i++) {
    float4 nv = np[i], v = xs[i], o;
    o.x = qdv(v.x, nv.x, mn, scale, inv);
    o.y = qdv(v.y, nv.y, mn, scale, inv);
    o.z = qdv(v.z, nv.z, mn, scale, inv);
    o.w = qdv(v.w, nv.w, mn, scale, inv);
    op[i] = o;
  }
}

// ---------------- fused quant-dequant + GEMM: Y(Nx16) = qd(relu?(X)) (NxK) @ W (Kx16) ----------------
// One wave32 computes a 16x16 output tile with V_WMMA_F32_16X16X4_F32, looping K in steps of 4.
template<int K, bool RELU>
__global__ void gemm_qd_kernel(const float* __restrict__ X, const float* __restrict__ NZ,
                               const float* __restrict__ W, float* __restrict__ Y,
                               int nRows) {
  int wave = blockIdx.x * (blockDim.x >> 5) + (threadIdx.x >> 5);
  int lane = threadIdx.x & 31;
  int nTiles = (nRows + 15) >> 4;
  if (wave >= nTiles) return;           // wave-uniform: EXEC stays all-1s for WMMA
  int r0 = wave << 4;
  int rowi = lane & 15;
  int half = lane >> 4;
  int r = r0 + rowi;
  const float* xr = X + (size_t)r * K;
  const float* nr = NZ + (size_t)r * K;

  // Pass A: per-row min/max (lanes l and l+16 each scan half of row l&15)
  float mn = 3.4e38f, mx = -3.4e38f;
#pragma unroll
  for (int j = 0; j < K / 2; j += 4) {
    float4 v = *reinterpret_cast<const float4*>(xr + half * (K / 2) + j);
    if (RELU) { v.x = fmaxf(v.x, 0.f); v.y = fmaxf(v.y, 0.f);
                v.z = fmaxf(v.z, 0.f); v.w = fmaxf(v.w, 0.f); }
    mn = fminf(mn, fminf(fminf(v.x, v.y), fminf(v.z, v.w)));
    mx = fmaxf(mx, fmaxf(fmaxf(v.x, v.y), fmaxf(v.z, v.w)));
  }
  mn = fminf(mn, __shfl_xor(mn, 16, 32));
  mx = fmaxf(mx, __shfl_xor(mx, 16, 32));
  float scale = QMAXF / (mx - mn);
  float inv = 1.0f / scale;

  // Pass B: K/4 WMMA steps. ISA f32 layout:
  //   A vgpr0 = K=k0   (lanes 0-15) / K=k0+2 (lanes 16-31); vgpr1 = k0+1 / k0+3
  //   B mirrored with lane = N.
  v8f c = {};
  for (int k0 = 0; k0 < K; k0 += 4) {
    int ka = k0 + half * 2;
    float2 xv = *reinterpret_cast<const float2*>(xr + ka);
    float2 nv = *reinterpret_cast<const float2*>(nr + ka);
    if (RELU) { xv.x = fmaxf(xv.x, 0.f); xv.y = fmaxf(xv.y, 0.f); }
    v2f a, b;
    a[0] = qdv(xv.x, nv.x, mn, scale, inv);
    a[1] = qdv(xv.y, nv.y, mn, scale, inv);
    b[0] = W[(ka + 0) * 16 + rowi];
    b[1] = W[(ka + 1) * 16 + rowi];
    c = wmma_f32_k4(a, b, c);
  }

  // C/D layout: vgpr v, lanes 0-15: (M=v, N=lane); lanes 16-31: (M=v+8, N=lane-16)
  if (r0 + 16 <= nRows) {               // wave-uniform fast path: unguarded stores
#pragma unroll
    for (int v = 0; v < 8; v++) {
      int rr = r0 + v + half * 8;
      Y[(size_t)rr * 16 + rowi] = c[v];
    }
  } else {
#pragma unroll
    for (int v = 0; v < 8; v++) {
      int rr = r0 + v + half * 8;
      if (rr < nRows) Y[(size_t)rr * 16 + rowi] = c[v];
    }
  }
}

// ---------------- final GEMM: O(Nx40) = Z(Nx16) @ W3(16x40), WMMA with masked N-tiles ----------------
__global__ void gemm40_kernel(const float* __restrict__ Z, const float* __restrict__ W,
                              float* __restrict__ O, int nRows) {
  int wave = blockIdx.x * (blockDim.x >> 5) + (threadIdx.x >> 5);
  int lane = threadIdx.x & 31;
  int tilesM = (nRows + 15) >> 4;
  int tm = wave / 3, tn = wave - tm * 3;   // 3 N-tiles cover 40 cols (last one partial)
  if (tm >= tilesM) return;                // wave-uniform
  int r0 = tm << 4, n0 = tn << 4;
  int rowi = lane & 15;
  int half = lane >> 4;
  const float* zr = Z + (size_t)(r0 + rowi) * 16;
  int cn = n0 + rowi;
  int ci = cn < 40 ? cn : 39;              // clamp address, select value (no EXEC games)
  v8f c = {};
#pragma unroll
  for (int k0 = 0; k0 < 16; k0 += 4) {
    int ka = k0 + half * 2;
    float2 av = *reinterpret_cast<const float2*>(zr + ka);
    v2f a, b;
    a[0] = av.x;
    a[1] = av.y;
    float b0 = W[(ka + 0) * 40 + ci];
    float b1 = W[(ka + 1) * 40 + ci];
    b[0] = (cn < 40) ? b0 : 0.0f;
    b[1] = (cn < 40) ? b1 : 0.0f;
    c = wmma_f32_k4(a, b, c);
  }
  if (r0 + 16 <= nRows) {                  // wave-uniform row guard hoisted
#pragma unroll
    for (int v = 0; v < 8; v++) {
      int rr = r0 + v + half * 8;
      if (cn < 40) O[(size_t)rr * 40 + cn] = c[v];
    }
  } else {
#pragma unroll
    for (int v = 0; v < 8; v++) {
      int rr = r0 + v + half * 8;
      if (rr < nRows && cn < 40) O[(size_t)rr * 40 + cn] = c[v];
    }
  }
}

extern "C" void kernel_launch(void* const* d_in, const int* in_sizes, int n_in,
                              void* d_out, int out_size, void* d_ws, size_t ws_size,
                              hipStream_t stream) {
  const float* features = (const float*)d_in[0];   // [N,128]
  const int*   row      = (const int*)d_in[1];     // [E]
  const int*   col      = (const int*)d_in[2];     // [E]
  const float* vals     = (const float*)d_in[3];   // [E]
  const float* W1       = (const float*)d_in[4];   // [128,16]
  const float* W2       = (const float*)d_in[5];   // [16,16]
  const float* W3       = (const float*)d_in[6];   // [16,40]
  const float* noise1   = (const float*)d_in[7];   // [N,128]
  const float* noise2   = (const float*)d_in[8];   // [N,16]
  const float* noise3   = (const float*)d_in[9];   // [N,16]

  int N = in_sizes[0] / 128;
  int E = in_sizes[1];

  float* bufA = (float*)d_ws;                       // N*16 floats
  float* bufB = bufA + (size_t)N * 16;              // N*16 floats
  float* bufC = bufB + (size_t)N * 16;              // N*16 floats
  float* outp = (float*)d_out;                      // [N,40]

  int tiles = (N + 15) / 16;
  dim3 blk(256);
  int gemmBlocks = (tiles + 7) / 8;                 // 8 waves per block
  int zeroBlocks = (N * 16 + 255) / 256;
  int spmmBlocks = (E * 4 + 255) / 256;

  // ----- layer 1: y1 = qd(x) @ W1 ; z1 = A @ y1  (reassociated: A@(dq@W) == (A@dq)@W)
  gemm_qd_kernel<128, false><<<gemmBlocks, blk, 0, stream>>>(features, noise1, W1, bufA, N);
  zero_kernel<<<zeroBlocks, blk, 0, stream>>>(bufB, N * 16);
  spmm16_kernel<<<spmmBlocks, blk, 0, stream>>>(row, col, vals, bufA, bufB, E);

  // ----- layer 2: y2 = qd(relu(z1)) @ W2 ; z2 = A @ y2
  gemm_qd_kernel<16, true><<<gemmBlocks, blk, 0, stream>>>(bufB, noise2, W2, bufC, N);
  zero_kernel<<<zeroBlocks, blk, 0, stream>>>(bufA, N * 16);
  spmm16_kernel<<<spmmBlocks, blk, 0, stream>>>(row, col, vals, bufC, bufA, E);

  // ----- layer 3: dq3 = qd(relu(z2)) ; z3 = A @ dq3 ; out = z3 @ W3
  qd16_kernel<true><<<(N + 255) / 256, blk, 0, stream>>>(bufA, noise3, bufB, N);
  zero_kernel<<<zeroBlocks, blk, 0, stream>>>(bufC, N * 16);
  spmm16_kernel<<<spmmBlocks, blk, 0, stream>>>(row, col, vals, bufB, bufC, E);
  gemm40_kernel<<<(tiles * 3 + 7) / 8, blk, 0, stream>>>(bufC, W3, outp, N);
}